// GCN_35613868819115
// MI455X (gfx1250) — compile-verified
//
#include <hip/hip_runtime.h>

typedef float v2f __attribute__((ext_vector_type(2)));
typedef float v8f __attribute__((ext_vector_type(8)));

// ---------------------------------------------------------------------------
// Small elementwise kernels
// ---------------------------------------------------------------------------
__global__ void fill_f32(float* p, float v, int n) {
    int i = blockIdx.x * blockDim.x + threadIdx.x;
    if (i < n) p[i] = v;
}

__global__ void deg_count(const long long* __restrict__ dst, float* __restrict__ deg, int E) {
    int e = blockIdx.x * blockDim.x + threadIdx.x;
    if (e < E) atomicAdd(&deg[(int)dst[e]], 1.0f);
}

__global__ void deg_rsqrt(float* d, int n) {
    int i = blockIdx.x * blockDim.x + threadIdx.x;
    if (i < n) d[i] = rsqrtf(d[i]);   // deg >= 1 always (self loops)
}

// hout[i,f] = hin[i,f]*dinv[i]^2 + bias[f]   (self-loop term + bias)
__global__ void agg_init(const float* __restrict__ hin, const float* __restrict__ dinv,
                         const float* __restrict__ bias, float* __restrict__ hout, int n) {
    int idx = blockIdx.x * blockDim.x + threadIdx.x;   // over n*128
    int row = idx >> 7;
    int f   = idx & 127;
    if (row < n) {
        float di = dinv[row];
        hout[idx] = hin[idx] * di * di + bias[f];
    }
}

// One wave per edge: gather 128 floats of src row (float4/lane), scaled atomic add to dst row
__global__ __launch_bounds__(256) void edge_scatter(
    const long long* __restrict__ src, const long long* __restrict__ dst,
    const float* __restrict__ dinv, const float* __restrict__ hin,
    float* __restrict__ hout, int E)
{
    int e    = blockIdx.x * 8 + (threadIdx.x >> 5);
    int lane = threadIdx.x & 31;
    if (e >= E) return;
    int s = (int)src[e];
    int d = (int)dst[e];
    float w = dinv[s] * dinv[d];
    const float4 v = ((const float4*)(hin + (size_t)s * 128))[lane];
    float* p = hout + (size_t)d * 128 + lane * 4;
    atomicAdd(p + 0, v.x * w);
    atomicAdd(p + 1, v.y * w);
    atomicAdd(p + 2, v.z * w);
    atomicAdd(p + 3, v.w * w);
}

// One wave per node: pooled[batch[i]] += relu(h[i])
__global__ __launch_bounds__(256) void pool_kernel(
    const float* __restrict__ h, const long long* __restrict__ batch,
    float* __restrict__ pooled, int n)
{
    int node = blockIdx.x * 8 + (threadIdx.x >> 5);
    int lane = threadIdx.x & 31;
    if (node >= n) return;
    int g = (int)batch[node];
    const float4 v = ((const float4*)(h + (size_t)node * 128))[lane];
    float* p = pooled + (size_t)g * 128 + lane * 4;
    atomicAdd(p + 0, fmaxf(v.x, 0.f));
    atomicAdd(p + 1, fmaxf(v.y, 0.f));
    atomicAdd(p + 2, fmaxf(v.z, 0.f));
    atomicAdd(p + 3, fmaxf(v.w, 0.f));
}

// ---------------------------------------------------------------------------
// fp32 WMMA GEMM: D[rows,NC] = (RELU_A ? relu(A) : A)[rows,128] @ W[128,NC] (+ bias)
// 128 threads = 4 waves; each wave owns a 16-row strip, all NC columns.
// V_WMMA_F32_16X16X4_F32 VGPR layout (wave32):
//   A (16x4): lanes 0-15 K={0,1}, lanes 16-31 K={2,3} of rows row0+ln
//   B (4x16): VGPR0 = K rows {0,2}, VGPR1 = K rows {1,3}, col = ln
//   C/D     : VGPR r -> row row0 + r + 8*half, col = ln
//
// W is staged into LDS pair-interleaved: element (k, c) lives at
//   sW[(k>>1)*(2*NC) + 2*c + (k&1)]
// so each B fragment (rows k, k+1 at one column) is ONE aligned ds_load_b64
// into an even VGPR pair -> no v_mov shuffling in front of v_wmma.
// ---------------------------------------------------------------------------
template <int NC, bool RELU_A, bool ADD_BIAS>
__global__ __launch_bounds__(128) void gemm_wmma(
    const float* __restrict__ A, const float* __restrict__ W,
    const float* __restrict__ bias, float* __restrict__ D, int rows)
{
    constexpr int K = 128;
    __shared__ __align__(16) float sW[K * NC];

    // stage W into LDS, pair-interleaved
    {
        const float4* Wv = (const float4*)W;
        for (int i = threadIdx.x; i < K * NC / 4; i += blockDim.x) {
            float4 v = Wv[i];
            int lin = i * 4;
            int k   = lin / NC;   // K row
            int c   = lin % NC;   // column (multiple of 4)
            float* base = sW + (k >> 1) * (2 * NC) + (k & 1);
            base[(c + 0) * 2] = v.x;
            base[(c + 1) * 2] = v.y;
            base[(c + 2) * 2] = v.z;
            base[(c + 3) * 2] = v.w;
        }
    }
    __syncthreads();

    const int wave = threadIdx.x >> 5;
    const int lane = threadIdx.x & 31;
    const int half = lane >> 4;     // 0: lanes 0-15, 1: lanes 16-31
    const int ln   = lane & 15;
    const int row0 = (blockIdx.x * 4 + wave) * 16;
    if (row0 >= rows) return;       // rows % 16 == 0, whole-wave uniform

    v8f acc[NC / 16];
#pragma unroll
    for (int t = 0; t < NC / 16; ++t) acc[t] = (v8f){0.f,0.f,0.f,0.f,0.f,0.f,0.f,0.f};

    const float* Arow = A + (size_t)(row0 + ln) * K;
    const v2f*   sWp  = (const v2f*)sW;   // v2f j -> floats {2j, 2j+1}

#pragma unroll 2
    for (int k0 = 0; k0 < K; k0 += 4) {
        float a0 = Arow[k0 + 2 * half + 0];
        float a1 = Arow[k0 + 2 * half + 1];
        if (RELU_A) { a0 = fmaxf(a0, 0.f); a1 = fmaxf(a1, 0.f); }
        v2f a; a.x = a0; a.y = a1;
        const int kp = (k0 >> 1) + half;   // K-pair index for this lane half
#pragma unroll
        for (int t = 0; t < NC / 16; ++t) {
            v2f b = sWp[kp * NC + t * 16 + ln];   // rows {k,k+1}, col t*16+ln
            acc[t] = __builtin_amdgcn_wmma_f32_16x16x4_f32(
                false, a, false, b, (short)0, acc[t], false, false);
        }
    }

#pragma unroll
    for (int t = 0; t < NC / 16; ++t) {
        if constexpr (ADD_BIAS) {
            float bv = bias[t * 16 + ln];
#pragma unroll
            for (int r = 0; r < 8; ++r)
                D[(size_t)(row0 + r + 8 * half) * NC + t * 16 + ln] = acc[t][r] + bv;
        } else {
#pragma unroll
            for (int r = 0; r < 8; ++r)
                D[(size_t)(row0 + r + 8 * half) * NC + t * 16 + ln] = acc[t][r];
        }
    }
}

// ---------------------------------------------------------------------------
// Host-side launch
// ---------------------------------------------------------------------------
extern "C" void kernel_launch(void* const* d_in, const int* in_sizes, int n_in,
                              void* d_out, int out_size, void* d_ws, size_t ws_size,
                              hipStream_t stream)
{
    const float*     x     = (const float*)d_in[0];
    const long long* eidx  = (const long long*)d_in[1];   // [2,E] int64
    const long long* batch = (const long long*)d_in[2];   // [N]   int64
    const float*     W1    = (const float*)d_in[3];
    const float*     b1    = (const float*)d_in[4];
    const float*     W2    = (const float*)d_in[5];
    const float*     b2    = (const float*)d_in[6];
    const float*     Wl    = (const float*)d_in[7];
    const float*     bl    = (const float*)d_in[8];

    const int N = in_sizes[2];
    const int E = in_sizes[1] / 2;
    const int G = 512;

    const long long* src = eidx;       // edge_index[0]
    const long long* dst = eidx + E;   // edge_index[1]

    // workspace layout (floats): dinv[N] | bufA[N*128] | bufB[N*128] | pooled[G*128]
    float* ws     = (float*)d_ws;
    float* dinv   = ws;
    float* bufA   = dinv + N;
    float* bufB   = bufA + (size_t)N * 128;
    float* pooled = bufB + (size_t)N * 128;

    const int T = 256;

    // degrees (self-loop contributes 1) -> dinv = rsqrt(deg)
    fill_f32<<<(N + T - 1) / T, T, 0, stream>>>(dinv, 1.0f, N);
    deg_count<<<(E + T - 1) / T, T, 0, stream>>>(dst, dinv, E);
    deg_rsqrt<<<(N + T - 1) / T, T, 0, stream>>>(dinv, N);

    // layer 1: bufA = x @ W1 ; bufB = dinv^2*bufA + b1 ; scatter edges
    gemm_wmma<128, false, false><<<(N + 63) / 64, 128, 0, stream>>>(x, W1, nullptr, bufA, N);
    agg_init<<<(N * 128 + T - 1) / T, T, 0, stream>>>(bufA, dinv, b1, bufB, N);
    edge_scatter<<<(E + 7) / 8, T, 0, stream>>>(src, dst, dinv, bufA, bufB, E);

    // layer 2: bufA = relu(bufB) @ W2 ; bufB = dinv^2*bufA + b2 ; scatter edges
    gemm_wmma<128, true, false><<<(N + 63) / 64, 128, 0, stream>>>(bufB, W2, nullptr, bufA, N);
    agg_init<<<(N * 128 + T - 1) / T, T, 0, stream>>>(bufA, dinv, b2, bufB, N);
    edge_scatter<<<(E + 7) / 8, T, 0, stream>>>(src, dst, dinv, bufA, bufB, E);

    // pool: pooled[g] = sum relu(bufB[i]) over batch[i]==g
    fill_f32<<<(G * 128 + T - 1) / T, T, 0, stream>>>(pooled, 0.0f, G * 128);
    pool_kernel<<<(N + 7) / 8, T, 0, stream>>>(bufB, batch, pooled, N);

    // head: out[512,64] = pooled @ Wl + bl
    gemm_wmma<64, false, true><<<(G + 63) / 64, 128, 0, stream>>>(pooled, Wl, bl, (float*)d_out, G);
}